// EdgeClassifierGNN_55551107006974
// MI455X (gfx1250) — compile-verified
//
#include <hip/hip_runtime.h>
#include <hip/hip_bf16.h>

// ---------------------------------------------------------------------------
// EdgeClassifierGNN for MI455X (gfx1250), wave32 + V_WMMA_F32_16X16X4_F32.
//
// Roofline: ~19.5 GFLOP fp32 total; gathered node-feature arrays (12.8 MB)
// are L2-resident (192 MB L2) -> L2-gather/atomic bound, not HBM bound.
// fp32 WMMA keeps reference numerics (RNE, fp32 accumulate).
// A-fragments are hoisted into registers (loaded once as b64) and reused
// across all N-tiles so each random h[src]/h[dst] row is gathered once.
// ---------------------------------------------------------------------------

typedef float v2f __attribute__((ext_vector_type(2)));
typedef float v8f __attribute__((ext_vector_type(8)));

__device__ __forceinline__ v8f wmma_f32x4(v2f a, v2f b, v8f c) {
  // D = A(16x4) * B(4x16) + C(16x16), fp32
  return __builtin_amdgcn_wmma_f32_16x16x4_f32(
      /*neg_a=*/false, a, /*neg_b=*/false, b,
      /*c_mod=*/(short)0, c, /*reuse_a=*/false, /*reuse_b=*/false);
}

__device__ __forceinline__ v2f load2(const float* p) {
  return *(const v2f*)p;  // 8B-aligned by construction (even k, aligned rows)
}

// ---------------------------------------------------------------------------
// Utility kernels
// ---------------------------------------------------------------------------
__global__ void zero_f32_kernel(float* __restrict__ p, int n) {
  int i = blockIdx.x * blockDim.x + threadIdx.x;
  if (i < n) p[i] = 0.0f;
}

__global__ void degree_kernel(const int* __restrict__ dst,
                              float* __restrict__ cnt, int E) {
  int e = blockIdx.x * blockDim.x + threadIdx.x;
  if (e < E) atomicAdd(&cnt[dst[e]], 1.0f);
}

// agg[dst] += x[src]; one thread handles 4 features of one edge.
template <int F>
__global__ void scatter_kernel(const float* __restrict__ x,
                               const int* __restrict__ src,
                               const int* __restrict__ dst,
                               float* __restrict__ agg, int E) {
  const int chunks = F / 4;
  int t = blockIdx.x * blockDim.x + threadIdx.x;
  int e = t / chunks;
  int c = t - e * chunks;
  if (e < E) {
    int s = src[e];
    int d = dst[e];
    const float4 v = ((const float4*)(x + (size_t)s * F))[c];
    float* ap = agg + (size_t)d * F + c * 4;
    atomicAdd(ap + 0, v.x);
    atomicAdd(ap + 1, v.y);
    atomicAdd(ap + 2, v.z);
    atomicAdd(ap + 3, v.w);
  }
}

// ---------------------------------------------------------------------------
// SAGE node update: hout = relu( (agg/max(cnt,1)) @ Wl + xin @ Wr + b )
// One wave per 16-node tile; FIN in {16,64}, FOUT = 64.
// A frag: lane = hi*16 + m -> A[m][k0+2hi], A[m][k0+2hi+1]
// B frag: lane = hi*16 + n -> B[k0+2hi][n], B[k0+2hi+1][n]
// C/D:    lane = hi*16 + n, vgpr r -> C[r+8hi][n]
// ---------------------------------------------------------------------------
template <int FIN>
__global__ __launch_bounds__(32) void sage_node_kernel(
    const float* __restrict__ xin,   // [N, FIN]
    const float* __restrict__ agg,   // [N, FIN]
    const float* __restrict__ cnt,   // [N]
    const float* __restrict__ Wl,    // [FIN, 64]
    const float* __restrict__ Wr,    // [FIN, 64]
    const float* __restrict__ bias,  // [64]
    float* __restrict__ hout,        // [N, 64]
    int N) {
  const int lane = threadIdx.x & 31;
  const int m = lane & 15;
  const int hi = lane >> 4;
  const int base = blockIdx.x * 16;

  int row = base + m;
  if (row >= N) row = N - 1;  // clamp loads; keep EXEC all-ones for WMMA
  const float inv = 1.0f / fmaxf(cnt[row], 1.0f);
  const float* arow = agg + (size_t)row * FIN;
  const float* xrow = xin + (size_t)row * FIN;

  // Hoist A fragments (loaded once, reused across 4 N-tiles).
  constexpr int KS = FIN / 4;
  v2f aM[KS], aX[KS];
#pragma unroll
  for (int i = 0; i < KS; ++i) {
    const int k = i * 4 + 2 * hi;
    v2f t = load2(arow + k);
    aM[i][0] = t[0] * inv;
    aM[i][1] = t[1] * inv;
    aX[i] = load2(xrow + k);
  }

#pragma unroll
  for (int nt = 0; nt < 4; ++nt) {
    const int ncol = nt * 16 + m;  // n == lane&15 for B/C/D frags
    v8f acc = (v8f){};
#pragma unroll
    for (int i = 0; i < KS; ++i) {
      const int k = i * 4 + 2 * hi;
      v2f b;
      b[0] = Wl[(size_t)k * 64 + ncol];
      b[1] = Wl[(size_t)(k + 1) * 64 + ncol];
      acc = wmma_f32x4(aM[i], b, acc);
    }
#pragma unroll
    for (int i = 0; i < KS; ++i) {
      const int k = i * 4 + 2 * hi;
      v2f b;
      b[0] = Wr[(size_t)k * 64 + ncol];
      b[1] = Wr[(size_t)(k + 1) * 64 + ncol];
      acc = wmma_f32x4(aX[i], b, acc);
    }
    // bias + relu + store (C/D layout)
    const float bc = bias[ncol];
#pragma unroll
    for (int r = 0; r < 8; ++r) {
      int orow = base + r + 8 * hi;
      if (orow < N) {
        hout[(size_t)orow * 64 + ncol] = fmaxf(acc[r] + bc, 0.0f);
      }
    }
  }
}

// ---------------------------------------------------------------------------
// Fused edge MLP: logits = ( relu( relu([h_s|h_d|ea] @ Wm1 + bm1) @ Wm2 + bm2 )
//                            @ Wm3 + bm3 )
// One wave per 16-edge tile; z1 (16x64) and z2 (16x32) staged through LDS.
// h[src]/h[dst] rows gathered exactly once into A-fragment registers.
// ---------------------------------------------------------------------------
__global__ __launch_bounds__(32) void edge_mlp_kernel(
    const float* __restrict__ h,      // [N, 64]  (h3)
    const int* __restrict__ src,      // [E]
    const int* __restrict__ dst,      // [E]
    const float* __restrict__ eattr,  // [E, 8]
    const float* __restrict__ Wm1,    // [136, 64]
    const float* __restrict__ bm1,    // [64]
    const float* __restrict__ Wm2,    // [64, 32]
    const float* __restrict__ bm2,    // [32]
    const float* __restrict__ Wm3,    // [32, 1]
    const float* __restrict__ bm3,    // [1]
    float* __restrict__ out,          // [E]
    int E) {
  __shared__ float z1buf[16 * 64];
  __shared__ float z2buf[16 * 32];

  const int lane = threadIdx.x & 31;
  const int m = lane & 15;
  const int hi = lane >> 4;
  const int base = blockIdx.x * 16;

  int e = base + m;
  if (e >= E) e = E - 1;  // clamp loads; EXEC stays all-ones for WMMA
  const int sn = src[e];
  const int dn = dst[e];
  const float* hs = h + (size_t)sn * 64;
  const float* hd = h + (size_t)dn * 64;
  const float* ea = eattr + (size_t)e * 8;

  // Hoist A fragments: 16 (h_src) + 16 (h_dst) + 2 (attr) float2 per lane.
  v2f aS[16], aD[16], aE[2];
#pragma unroll
  for (int i = 0; i < 16; ++i) {
    const int k = i * 4 + 2 * hi;
    aS[i] = load2(hs + k);
    aD[i] = load2(hd + k);
  }
#pragma unroll
  for (int i = 0; i < 2; ++i) {
    aE[i] = load2(ea + i * 4 + 2 * hi);
  }

  // ---- GEMM1: [16 x 136] @ [136 x 64] ----
#pragma unroll 1
  for (int nt = 0; nt < 4; ++nt) {
    const int ncol = nt * 16 + m;
    v8f acc = (v8f){};
#pragma unroll
    for (int i = 0; i < 16; ++i) {  // Wm1 rows 0..63 <- h[src]
      const int k = i * 4 + 2 * hi;
      v2f b;
      b[0] = Wm1[(size_t)k * 64 + ncol];
      b[1] = Wm1[(size_t)(k + 1) * 64 + ncol];
      acc = wmma_f32x4(aS[i], b, acc);
    }
#pragma unroll
    for (int i = 0; i < 16; ++i) {  // Wm1 rows 64..127 <- h[dst]
      const int k = 64 + i * 4 + 2 * hi;
      v2f b;
      b[0] = Wm1[(size_t)k * 64 + ncol];
      b[1] = Wm1[(size_t)(k + 1) * 64 + ncol];
      acc = wmma_f32x4(aD[i], b, acc);
    }
#pragma unroll
    for (int i = 0; i < 2; ++i) {   // Wm1 rows 128..135 <- edge_attr
      const int k = 128 + i * 4 + 2 * hi;
      v2f b;
      b[0] = Wm1[(size_t)k * 64 + ncol];
      b[1] = Wm1[(size_t)(k + 1) * 64 + ncol];
      acc = wmma_f32x4(aE[i], b, acc);
    }
    const float bc = bm1[ncol];
#pragma unroll
    for (int r = 0; r < 8; ++r) {
      z1buf[(r + 8 * hi) * 64 + ncol] = fmaxf(acc[r] + bc, 0.0f);
    }
  }
  __syncthreads();

  // ---- GEMM2: [16 x 64] @ [64 x 32] ----
  v2f aZ[16];
#pragma unroll
  for (int i = 0; i < 16; ++i) {
    aZ[i] = load2(&z1buf[m * 64 + i * 4 + 2 * hi]);
  }
#pragma unroll 1
  for (int nt = 0; nt < 2; ++nt) {
    const int ncol = nt * 16 + m;
    v8f acc = (v8f){};
#pragma unroll
    for (int i = 0; i < 16; ++i) {
      const int k = i * 4 + 2 * hi;
      v2f b;
      b[0] = Wm2[(size_t)k * 32 + ncol];
      b[1] = Wm2[(size_t)(k + 1) * 32 + ncol];
      acc = wmma_f32x4(aZ[i], b, acc);
    }
    const float bc = bm2[ncol];
#pragma unroll
    for (int r = 0; r < 8; ++r) {
      z2buf[(r + 8 * hi) * 32 + ncol] = fmaxf(acc[r] + bc, 0.0f);
    }
  }
  __syncthreads();

  // ---- GEMM3: [16 x 32] @ [32 x 1] (scalar dot, lanes 0..15) ----
  if (lane < 16) {
    const int eo = base + lane;
    if (eo < E) {
      float s = bm3[0];
#pragma unroll
      for (int k = 0; k < 32; ++k) s += z2buf[lane * 32 + k] * Wm3[k];
      out[eo] = s;
    }
  }
}

// ---------------------------------------------------------------------------
// Host launcher
// ---------------------------------------------------------------------------
extern "C" void kernel_launch(void* const* d_in, const int* in_sizes, int n_in,
                              void* d_out, int out_size, void* d_ws,
                              size_t ws_size, hipStream_t stream) {
  const float* x     = (const float*)d_in[0];   // [N,16]
  const int*   ei    = (const int*)d_in[1];     // [2,E]
  const float* eattr = (const float*)d_in[2];   // [E,8]
  const float* W1l = (const float*)d_in[3];
  const float* W1r = (const float*)d_in[4];
  const float* b1  = (const float*)d_in[5];
  const float* W2l = (const float*)d_in[6];
  const float* W2r = (const float*)d_in[7];
  const float* b2  = (const float*)d_in[8];
  const float* W3l = (const float*)d_in[9];
  const float* W3r = (const float*)d_in[10];
  const float* b3  = (const float*)d_in[11];
  const float* Wm1 = (const float*)d_in[12];
  const float* bm1 = (const float*)d_in[13];
  const float* Wm2 = (const float*)d_in[14];
  const float* bm2 = (const float*)d_in[15];
  const float* Wm3 = (const float*)d_in[16];
  const float* bm3 = (const float*)d_in[17];
  float* out = (float*)d_out;

  const int N = in_sizes[0] / 16;
  const int E = in_sizes[1] / 2;
  const int* src = ei;
  const int* dst = ei + E;

  // Workspace: cnt[N] | agg[N*64] | hA[N*64] | hB[N*64]  (~38.6 MB)
  float* cnt = (float*)d_ws;
  float* agg = cnt + N;
  float* hA = agg + (size_t)N * 64;
  float* hB = hA + (size_t)N * 64;

  const int ZT = 256;
  auto blks = [](long long n, int t) { return (unsigned)((n + t - 1) / t); };
  const unsigned nodeTiles = blks(N, 16);
  const unsigned edgeTiles = blks(E, 16);

  // degree counts (shared by all 3 layers)
  zero_f32_kernel<<<blks(N, ZT), ZT, 0, stream>>>(cnt, N);
  degree_kernel<<<blks(E, ZT), ZT, 0, stream>>>(dst, cnt, E);

  // ---- Layer 1: IN=16 -> 64 ----
  zero_f32_kernel<<<blks((long long)N * 16, ZT), ZT, 0, stream>>>(agg, N * 16);
  scatter_kernel<16><<<blks((long long)E * 4, ZT), ZT, 0, stream>>>(
      x, src, dst, agg, E);
  sage_node_kernel<16><<<nodeTiles, 32, 0, stream>>>(
      x, agg, cnt, W1l, W1r, b1, hA, N);

  // ---- Layer 2: 64 -> 64 ----
  zero_f32_kernel<<<blks((long long)N * 64, ZT), ZT, 0, stream>>>(agg, N * 64);
  scatter_kernel<64><<<blks((long long)E * 16, ZT), ZT, 0, stream>>>(
      hA, src, dst, agg, E);
  sage_node_kernel<64><<<nodeTiles, 32, 0, stream>>>(
      hA, agg, cnt, W2l, W2r, b2, hB, N);

  // ---- Layer 3: 64 -> 64 (write back into hA) ----
  zero_f32_kernel<<<blks((long long)N * 64, ZT), ZT, 0, stream>>>(agg, N * 64);
  scatter_kernel<64><<<blks((long long)E * 16, ZT), ZT, 0, stream>>>(
      hB, src, dst, agg, E);
  sage_node_kernel<64><<<nodeTiles, 32, 0, stream>>>(
      hB, agg, cnt, W3l, W3r, b3, hA, N);

  // ---- Fused edge MLP ----
  edge_mlp_kernel<<<edgeTiles, 32, 0, stream>>>(
      hA, src, dst, eattr, Wm1, bm1, Wm2, bm2, Wm3, bm3, out, E);
}